// GCN_71897752535696
// MI455X (gfx1250) — compile-verified
//
#include <hip/hip_runtime.h>
#include <hip/hip_bf16.h>

// ---------------------------------------------------------------------------
// 3-layer GraphSAGE (mean aggregation) for MI455X / gfx1250, wave32.
//
//   per layer:  agg[i]  = sum_{e: dst[e]==i} x[src[e]]        (f32 atomics)
//               mean[i] = agg[i] / max(deg[i], 1)
//               h[i]    = mean[i] @ Wl + b + x[i] @ Wr         (f32 WMMA)
//               relu (layers 1,2 only)
//
// Bottleneck is the edge scatter (~7.4 GB traffic + atomic RMW across the 3
// layers); total GEMM work is only ~26 GFLOP, so GEMMs run as exact-f32
// V_WMMA_F32_16X16X4_F32 (numerics match the f32 reference). Each block
// stages its 16 shared A-rows (mean & x) into LDS once (coalesced float4,
// inv-scaling folded in), then 8 waves each compute one 16x16 output tile
// from LDS fragments. Weights stay in L2 (<=256 KB vs 192 MB L2).
// ---------------------------------------------------------------------------

typedef float v2f __attribute__((ext_vector_type(2)));
typedef float v8f __attribute__((ext_vector_type(8)));

// ---- degree count: one thread per edge --------------------------------------
__global__ void sage_deg_count(const int* __restrict__ dst, float* __restrict__ cnt,
                               int nedges) {
    int e = blockIdx.x * blockDim.x + threadIdx.x;
    if (e >= nedges) return;
    atomicAdd(&cnt[dst[e]], 1.0f);
}

// ---- cnt -> 1/max(cnt,1), in place ------------------------------------------
__global__ void sage_inv_deg(float* __restrict__ cnt, int n) {
    int i = blockIdx.x * blockDim.x + threadIdx.x;
    if (i >= n) return;
    cnt[i] = 1.0f / fmaxf(cnt[i], 1.0f);
}

// ---- scatter-add: one thread per (edge, 4-feature chunk) --------------------
// Consecutive lanes cover consecutive 16B chunks of one source row (coalesced
// reads); the adds are dword global_atomic_add_f32 (no packed f32 atomic).
__global__ void sage_scatter_add(const float* __restrict__ xin,
                                 const int* __restrict__ src,
                                 const int* __restrict__ dst,
                                 float* __restrict__ agg,
                                 int f4, long long nwork) {
    long long tid = (long long)blockIdx.x * blockDim.x + threadIdx.x;
    if (tid >= nwork) return;
    int e = (int)(tid / f4);
    int c = (int)(tid % f4);
    int F = f4 * 4;
    int s = src[e];
    int d = dst[e];
    const float4 v = *(const float4*)(xin + (size_t)s * F + c * 4);
    float* p = agg + (size_t)d * F + c * 4;
    atomicAdd(p + 0, v.x);
    atomicAdd(p + 1, v.y);
    atomicAdd(p + 2, v.z);
    atomicAdd(p + 3, v.w);
}

// ---- fused SAGE GEMM:  out = (agg * inv) @ Wl + xin @ Wr + b  [, relu] ------
// Block = 256 threads = 8 waves. One block handles rows [tm*16, tm*16+16) and
// 8 consecutive 16-wide column tiles (wave w -> tile tn = cb*8 + w).
// LDS: sMean / sX, 16 rows x K floats, row stride K+4 (bank spread + 16B
// alignment). WMMA fragments per ISA layout:
//   A (16x4 f32): lane<16 -> K=k0,k0+1 of row (lane&15); lane>=16 -> k0+2,k0+3
//   B (4x16):     symmetric over N; C/D: VGPR r -> row r+8*(lane>=16), col lane&15
__global__ void __launch_bounds__(256)
sage_gemm_wmma(const float* __restrict__ agg, const float* __restrict__ inv,
               const float* __restrict__ xin,
               const float* __restrict__ Wl, const float* __restrict__ Wr,
               const float* __restrict__ bias, float* __restrict__ out,
               int K, int Nout, int nbx, int do_relu) {
    extern __shared__ float smem[];
    const int stride = K + 4;
    float* sMean = smem;                 // 16 x stride
    float* sX    = smem + 16 * stride;   // 16 x stride

    const int tm = blockIdx.x / nbx;     // row-tile of this block
    const int cb = blockIdx.x % nbx;     // column-block (8 tiles each)

    // ---- stage 16 rows of mean (= agg * inv) and x into LDS, coalesced ----
    const int total = 16 * K;
    for (int idx = threadIdx.x * 4; idx < total; idx += blockDim.x * 4) {
        const int r  = idx / K;
        const int c0 = idx % K;
        const size_t g = (size_t)(tm * 16 + r) * K + c0;
        const float iv = inv[tm * 16 + r];
        float4 va = *(const float4*)(agg + g);
        va.x *= iv; va.y *= iv; va.z *= iv; va.w *= iv;
        *(float4*)(sMean + r * stride + c0) = va;
        *(float4*)(sX + r * stride + c0) = *(const float4*)(xin + g);
    }
    __syncthreads();

    const int lane = threadIdx.x & 31;
    const int wave = threadIdx.x >> 5;
    const int m  = lane & 15;
    const int hi = lane >> 4;                   // 0 or 1
    const int tn = cb * 8 + wave;
    const int ncol = tn * 16 + m;               // output column this lane owns
    const float* sMrow = sMean + m * stride;
    const float* sXrow = sX + m * stride;

    v8f c = {0.f, 0.f, 0.f, 0.f, 0.f, 0.f, 0.f, 0.f};

    for (int k0 = 0; k0 < K; k0 += 4) {
        const int ka = k0 + 2 * hi;
        // mean path
        v2f a = *(const v2f*)(sMrow + ka);
        v2f b;
        b.x = Wl[(size_t)ka * Nout + ncol];
        b.y = Wl[(size_t)(ka + 1) * Nout + ncol];
        c = __builtin_amdgcn_wmma_f32_16x16x4_f32(false, a, false, b,
                                                  (short)0, c, false, false);
        // root path
        v2f a2 = *(const v2f*)(sXrow + ka);
        v2f b2;
        b2.x = Wr[(size_t)ka * Nout + ncol];
        b2.y = Wr[(size_t)(ka + 1) * Nout + ncol];
        c = __builtin_amdgcn_wmma_f32_16x16x4_f32(false, a2, false, b2,
                                                  (short)0, c, false, false);
    }

    const float bb = bias[ncol];
    #pragma unroll
    for (int r = 0; r < 8; ++r) {
        const int row = tm * 16 + r + 8 * hi;
        float v = c[r] + bb;
        if (do_relu) v = fmaxf(v, 0.0f);
        out[(size_t)row * Nout + ncol] = v;
    }
}

// ---------------------------------------------------------------------------
extern "C" void kernel_launch(void* const* d_in, const int* in_sizes, int n_in,
                              void* d_out, int out_size, void* d_ws, size_t ws_size,
                              hipStream_t stream) {
    const float* x    = (const float*)d_in[0];
    const int*   edge = (const int*)d_in[1];
    const float* W1l  = (const float*)d_in[2];
    const float* b1   = (const float*)d_in[3];
    const float* W1r  = (const float*)d_in[4];
    const float* W2l  = (const float*)d_in[5];
    const float* b2   = (const float*)d_in[6];
    const float* W2r  = (const float*)d_in[7];
    const float* W3l  = (const float*)d_in[8];
    const float* b3   = (const float*)d_in[9];
    const float* W3r  = (const float*)d_in[10];

    const int HIDDEN = in_sizes[3];             // 256 (len b1)
    const int OUTF   = in_sizes[9];             // 128 (len b3)
    const int D_IN   = in_sizes[2] / HIDDEN;    // 128
    const int M      = in_sizes[0] / D_IN;      // 50000 nodes
    const int E      = in_sizes[1] / 2;         // 800000 edges
    const int* src   = edge;
    const int* dst   = edge + E;

    float* w = (float*)d_ws;
    const size_t aggOff = ((size_t)M + 127) & ~(size_t)127;
    const size_t h1Off  = aggOff + (size_t)M * HIDDEN;
    const size_t h2Off  = h1Off  + (size_t)M * HIDDEN;
    float* inv = w;
    float* agg = w + aggOff;
    float* h1  = w + h1Off;
    float* h2  = w + h2Off;
    float* outp = (float*)d_out;

    const int B = 256;

    // degrees -> 1/max(deg,1)
    hipMemsetAsync(inv, 0, (size_t)M * sizeof(float), stream);
    sage_deg_count<<<(E + B - 1) / B, B, 0, stream>>>(dst, inv, E);
    sage_inv_deg<<<(M + B - 1) / B, B, 0, stream>>>(inv, M);

    // ---- layer 1: x(128) -> h1(256), relu ----
    {
        const int K = D_IN, N = HIDDEN, f4 = K / 4;
        hipMemsetAsync(agg, 0, (size_t)M * K * sizeof(float), stream);
        long long nw = (long long)E * f4;
        sage_scatter_add<<<(int)((nw + B - 1) / B), B, 0, stream>>>(x, src, dst, agg, f4, nw);
        const int nbx = (N / 16) / 8;
        const size_t smem = 2u * 16u * (size_t)(K + 4) * sizeof(float);
        sage_gemm_wmma<<<(M / 16) * nbx, 256, smem, stream>>>(agg, inv, x, W1l, W1r, b1,
                                                              h1, K, N, nbx, 1);
    }
    // ---- layer 2: h1(256) -> h2(256), relu ----
    {
        const int K = HIDDEN, N = HIDDEN, f4 = K / 4;
        hipMemsetAsync(agg, 0, (size_t)M * K * sizeof(float), stream);
        long long nw = (long long)E * f4;
        sage_scatter_add<<<(int)((nw + B - 1) / B), B, 0, stream>>>(h1, src, dst, agg, f4, nw);
        const int nbx = (N / 16) / 8;
        const size_t smem = 2u * 16u * (size_t)(K + 4) * sizeof(float);
        sage_gemm_wmma<<<(M / 16) * nbx, 256, smem, stream>>>(agg, inv, h1, W2l, W2r, b2,
                                                              h2, K, N, nbx, 1);
    }
    // ---- layer 3: h2(256) -> out(128), no relu ----
    {
        const int K = HIDDEN, N = OUTF, f4 = K / 4;
        hipMemsetAsync(agg, 0, (size_t)M * K * sizeof(float), stream);
        long long nw = (long long)E * f4;
        sage_scatter_add<<<(int)((nw + B - 1) / B), B, 0, stream>>>(h2, src, dst, agg, f4, nw);
        const int nbx = (N / 16) / 8;
        const size_t smem = 2u * 16u * (size_t)(K + 4) * sizeof(float);
        sage_gemm_wmma<<<(M / 16) * nbx, 256, smem, stream>>>(agg, inv, h2, W3l, W3r, b3,
                                                              outp, K, N, nbx, 0);
    }
}